// SAGE_26560077759041
// MI455X (gfx1250) — compile-verified
//
#include <hip/hip_runtime.h>

typedef __attribute__((ext_vector_type(2))) float v2f;
typedef __attribute__((ext_vector_type(8))) float v8f;

#define N_NODES 100000
#define N_EDGES 1600000
#define KDIM    128
#define NFEAT   ((size_t)N_NODES * KDIM)   // 12,800,000 floats

// ---------------------------------------------------------------------------
// zero a float buffer (grid-stride)
// ---------------------------------------------------------------------------
__global__ void zero_kernel(float* __restrict__ p, long n) {
  long i = (long)blockIdx.x * blockDim.x + threadIdx.x;
  long stride = (long)gridDim.x * blockDim.x;
  for (; i < n; i += stride) p[i] = 0.0f;
}

// ---------------------------------------------------------------------------
// in-degree count: deg[dst[e]] += 1.0
// ---------------------------------------------------------------------------
__global__ void degree_kernel(const int* __restrict__ dst, float* __restrict__ deg,
                              int nE) {
  int e = blockIdx.x * blockDim.x + threadIdx.x;
  if (e < nE)
    __hip_atomic_fetch_add(&deg[dst[e]], 1.0f, __ATOMIC_RELAXED,
                           __HIP_MEMORY_SCOPE_AGENT);
}

// ---------------------------------------------------------------------------
// deg -> 1/max(deg,1)
// ---------------------------------------------------------------------------
__global__ void invdeg_kernel(float* __restrict__ deg, int n) {
  int i = blockIdx.x * blockDim.x + threadIdx.x;
  if (i < n) deg[i] = 1.0f / fmaxf(deg[i], 1.0f);
}

// ---------------------------------------------------------------------------
// edge scatter: agg[dst[e], :] += h[src[e], :]
// one wave (32 lanes) per edge; each lane moves a float4 (128 floats/edge).
// Gather is a coalesced 512B burst; scatter-adds resolve in L2
// (51MB agg << 192MB L2), which is the roofline-limiting stage here.
// ---------------------------------------------------------------------------
__global__ void scatter_kernel(const float* __restrict__ h,
                               const int* __restrict__ src,
                               const int* __restrict__ dst,
                               float* __restrict__ agg, int nE) {
  int t = blockIdx.x * blockDim.x + threadIdx.x;
  int e = t >> 5;
  int lane = t & 31;
  if (e >= nE) return;
  int s = src[e];                     // uniform within the wave
  int d = dst[e];
  const float4 v = *((const float4*)(h + (size_t)s * KDIM) + lane);
  float* ap = agg + (size_t)d * KDIM + lane * 4;
  __hip_atomic_fetch_add(ap + 0, v.x, __ATOMIC_RELAXED, __HIP_MEMORY_SCOPE_AGENT);
  __hip_atomic_fetch_add(ap + 1, v.y, __ATOMIC_RELAXED, __HIP_MEMORY_SCOPE_AGENT);
  __hip_atomic_fetch_add(ap + 2, v.z, __ATOMIC_RELAXED, __HIP_MEMORY_SCOPE_AGENT);
  __hip_atomic_fetch_add(ap + 3, v.w, __ATOMIC_RELAXED, __HIP_MEMORY_SCOPE_AGENT);
}

// ---------------------------------------------------------------------------
// fused SAGE layer GEMM:
//   out = relu?( h @ Wself + (agg * invdeg) @ Wneigh + bias )
// one wave = one 32x16 output tile (two 16x16 accumulators sharing each B
// fragment) via V_WMMA_F32_16X16X4_F32, K=128.
// grid.x = M/32 row tiles; blockDim.x = ceil(NOUT/16) waves * 32.
// NOUT is a compile-time constant -> all W loads are base + immediate offset.
//
// f32 WMMA VGPR layouts (ISA 7.12.2):
//   A 16x4 : lane L -> m=L%16, VGPR v -> k = 2*(L/16)+v
//   B 4x16 : lane L -> n=L%16, VGPR v -> k = 2*(L/16)+v
//   C 16x16: lane L -> n=L%16, VGPR v -> m = v + 8*(L/16)
// ---------------------------------------------------------------------------
template <int NOUT, bool RELU>
__global__ void sage_gemm_kernel(const float* __restrict__ h,
                                 const float* __restrict__ agg,
                                 const float* __restrict__ invdeg,
                                 const float* __restrict__ Wself,
                                 const float* __restrict__ Wneigh,
                                 const float* __restrict__ bias,
                                 float* __restrict__ out) {
  constexpr bool FULL = (NOUT % 16 == 0);   // no partial N tile
  const int lane = threadIdx.x & 31;
  const int wave = threadIdx.x >> 5;
  const int mbase = blockIdx.x << 5;        // 32 rows per block (100000 = 3125*32)
  const int nbase = wave << 4;
  const int lm = lane & 15;
  const int half = lane >> 4;               // 0 or 1

  const int row0 = mbase + lm;
  const int row1 = row0 + 16;
  const int n = nbase + lm;
  const int nW = FULL ? n : ((n < NOUT) ? n : (NOUT - 1));
  const float nmask = (FULL || n < NOUT) ? 1.0f : 0.0f;

  // per-lane base pointers; loop offsets are compile-time immediates
  const float* a0 = h   + (size_t)row0 * KDIM + 2 * half;
  const float* a1 = h   + (size_t)row1 * KDIM + 2 * half;
  const float* g0 = agg + (size_t)row0 * KDIM + 2 * half;
  const float* g1 = agg + (size_t)row1 * KDIM + 2 * half;
  const float* ws = Wself  + (size_t)(2 * half) * NOUT + nW;
  const float* wn = Wneigh + (size_t)(2 * half) * NOUT + nW;
  const float idg0 = invdeg[row0];
  const float idg1 = invdeg[row1];

  v8f acc0 = {0.f, 0.f, 0.f, 0.f, 0.f, 0.f, 0.f, 0.f};
  v8f acc1 = {0.f, 0.f, 0.f, 0.f, 0.f, 0.f, 0.f, 0.f};

  // self term: h @ Wself  (one B fragment feeds two WMMAs)
#pragma unroll
  for (int k = 0; k < KDIM; k += 4) {
    v2f b;
    b.x = ws[k * NOUT];
    b.y = ws[k * NOUT + NOUT];
    if constexpr (!FULL) { b.x *= nmask; b.y *= nmask; }
    v2f av0 = *(const v2f*)(a0 + k);
    v2f av1 = *(const v2f*)(a1 + k);
    acc0 = __builtin_amdgcn_wmma_f32_16x16x4_f32(false, av0, false, b,
                                                 (short)0, acc0, false, false);
    acc1 = __builtin_amdgcn_wmma_f32_16x16x4_f32(false, av1, false, b,
                                                 (short)0, acc1, false, false);
  }

  // neighbor term: (agg * invdeg) @ Wneigh
#pragma unroll
  for (int k = 0; k < KDIM; k += 4) {
    v2f b;
    b.x = wn[k * NOUT];
    b.y = wn[k * NOUT + NOUT];
    if constexpr (!FULL) { b.x *= nmask; b.y *= nmask; }
    v2f av0 = *(const v2f*)(g0 + k);
    av0.x *= idg0; av0.y *= idg0;
    v2f av1 = *(const v2f*)(g1 + k);
    av1.x *= idg1; av1.y *= idg1;
    acc0 = __builtin_amdgcn_wmma_f32_16x16x4_f32(false, av0, false, b,
                                                 (short)0, acc0, false, false);
    acc1 = __builtin_amdgcn_wmma_f32_16x16x4_f32(false, av1, false, b,
                                                 (short)0, acc1, false, false);
  }

  if (FULL || n < NOUT) {
    const float bv = bias[n];
#pragma unroll
    for (int v = 0; v < 8; ++v) {
      const int m0 = mbase + v + 8 * half;
      float val0 = acc0[v] + bv;
      float val1 = acc1[v] + bv;
      if constexpr (RELU) { val0 = fmaxf(val0, 0.0f); val1 = fmaxf(val1, 0.0f); }
      out[(size_t)m0 * NOUT + n] = val0;
      out[(size_t)(m0 + 16) * NOUT + n] = val1;
    }
  }
}

// ---------------------------------------------------------------------------
// launcher
// ---------------------------------------------------------------------------
extern "C" void kernel_launch(void* const* d_in, const int* in_sizes, int n_in,
                              void* d_out, int out_size, void* d_ws, size_t ws_size,
                              hipStream_t stream) {
  const float* x   = (const float*)d_in[0];
  const int*   src = (const int*)  d_in[1];
  const int*   dst = (const int*)  d_in[2];
  const float* Ws0 = (const float*)d_in[3];
  const float* Wn0 = (const float*)d_in[4];
  const float* b0  = (const float*)d_in[5];
  const float* Ws1 = (const float*)d_in[6];
  const float* Wn1 = (const float*)d_in[7];
  const float* b1  = (const float*)d_in[8];
  const float* Ws2 = (const float*)d_in[9];
  const float* Wn2 = (const float*)d_in[10];
  const float* b2  = (const float*)d_in[11];
  float* out = (float*)d_out;

  // workspace layout (floats): invdeg | agg | h1 | h2   (~147 MB total)
  float* ws     = (float*)d_ws;
  float* invdeg = ws;                        // N_NODES floats
  float* agg    = ws + 100352;               // padded past invdeg
  float* h1     = agg + NFEAT;
  float* h2     = h1 + NFEAT;

  const int ZB = 2048, ZT = 256;
  const int degBlocks  = (N_EDGES + 255) / 256;
  const int nodeBlocks = (N_NODES + 255) / 256;
  const int scatBlocks = (int)(((long)N_EDGES * 32 + 255) / 256);
  const int mTiles = N_NODES / 32;           // 3125, exact

  // degrees -> 1/max(deg,1)
  zero_kernel<<<ZB, ZT, 0, stream>>>(invdeg, N_NODES);
  degree_kernel<<<degBlocks, 256, 0, stream>>>(dst, invdeg, N_EDGES);
  invdeg_kernel<<<nodeBlocks, 256, 0, stream>>>(invdeg, N_NODES);

  // layer 0: x -> h1 (relu)
  zero_kernel<<<ZB, ZT, 0, stream>>>(agg, (long)NFEAT);
  scatter_kernel<<<scatBlocks, 256, 0, stream>>>(x, src, dst, agg, N_EDGES);
  sage_gemm_kernel<128, true><<<mTiles, 256, 0, stream>>>(x, agg, invdeg,
                                                          Ws0, Wn0, b0, h1);

  // layer 1: h1 -> h2 (relu)
  zero_kernel<<<ZB, ZT, 0, stream>>>(agg, (long)NFEAT);
  scatter_kernel<<<scatBlocks, 256, 0, stream>>>(h1, src, dst, agg, N_EDGES);
  sage_gemm_kernel<128, true><<<mTiles, 256, 0, stream>>>(h1, agg, invdeg,
                                                          Ws1, Wn1, b1, h2);

  // layer 2: h2 -> out (47 classes, no relu); 3 waves cover 47 cols
  zero_kernel<<<ZB, ZT, 0, stream>>>(agg, (long)NFEAT);
  scatter_kernel<<<scatBlocks, 256, 0, stream>>>(h2, src, dst, agg, N_EDGES);
  sage_gemm_kernel<47, false><<<mTiles, 96, 0, stream>>>(h2, agg, invdeg,
                                                         Ws2, Wn2, b2, out);
}